// SignalNoiseRatio_87153476371100
// MI455X (gfx1250) — compile-verified
//
#include <hip/hip_runtime.h>
#include <math.h>

typedef float v2f __attribute__((ext_vector_type(2)));
typedef float v4f __attribute__((ext_vector_type(4)));
typedef float v8f __attribute__((ext_vector_type(8)));

#define EPS_F    1e-8f
#define B_ROWS   64
#define L_LEN    262144
#define SEGS     16
#define SEG_LEN  (L_LEN / SEGS)      // 16384 floats per segment
#define TPB      256                 // 8 wave32 per block
#define NWAVES   (TPB / 32)

// ---------------------------------------------------------------------------
// Kernel 1: streaming partial reductions.
// grid = (SEGS, B_ROWS); block (s = blockIdx.x) covers SEG_LEN contiguous
// floats of row b = blockIdx.y in each of the three inputs.
// Emits non-temporal global_load_b128 (data is touched exactly once; 192 MB
// total, don't pollute caches), per-lane FMA accumulation, wave32 shfl-xor
// tree, LDS cross-wave step. Writes (pt, tt, nn) partial per (row, seg).
// ---------------------------------------------------------------------------
__global__ __launch_bounds__(TPB)
void snr_partial_kernel(const float* __restrict__ t,
                        const float* __restrict__ p,
                        const float* __restrict__ w,
                        float* __restrict__ ws)
{
    const int s   = blockIdx.x;
    const int b   = blockIdx.y;
    const int tid = threadIdx.x;

    const size_t base = (size_t)b * L_LEN + (size_t)s * SEG_LEN;
    const v4f* __restrict__ t4 = (const v4f*)(t + base);
    const v4f* __restrict__ p4 = (const v4f*)(p + base);
    const v4f* __restrict__ w4 = (const v4f*)(w + base);

    float pt = 0.0f, tt = 0.0f, nn = 0.0f;

    // SEG_LEN/4 = 4096 float4 per segment -> 16 iterations per thread.
    constexpr int ITERS = (SEG_LEN / 4) / TPB;
#pragma unroll 4
    for (int k = 0; k < ITERS; ++k) {
        const int i = tid + k * TPB;
        v4f tv = __builtin_nontemporal_load(t4 + i);
        v4f pv = __builtin_nontemporal_load(p4 + i);
        v4f wv = __builtin_nontemporal_load(w4 + i);
        pt = fmaf(pv.x, tv.x, pt); pt = fmaf(pv.y, tv.y, pt);
        pt = fmaf(pv.z, tv.z, pt); pt = fmaf(pv.w, tv.w, pt);
        tt = fmaf(tv.x, tv.x, tt); tt = fmaf(tv.y, tv.y, tt);
        tt = fmaf(tv.z, tv.z, tt); tt = fmaf(tv.w, tv.w, tt);
        nn = fmaf(wv.x, wv.x, nn); nn = fmaf(wv.y, wv.y, nn);
        nn = fmaf(wv.z, wv.z, nn); nn = fmaf(wv.w, wv.w, nn);
    }

    // wave32 reduction
#pragma unroll
    for (int off = 16; off > 0; off >>= 1) {
        pt += __shfl_xor(pt, off, 32);
        tt += __shfl_xor(tt, off, 32);
        nn += __shfl_xor(nn, off, 32);
    }

    __shared__ float red[NWAVES * 3];
    const int wave = tid >> 5;
    const int lane = tid & 31;
    if (lane == 0) {
        red[wave * 3 + 0] = pt;
        red[wave * 3 + 1] = tt;
        red[wave * 3 + 2] = nn;
    }
    __syncthreads();

    if (tid < 32) {
        float a = 0.0f, c = 0.0f, d = 0.0f;
        if (tid < NWAVES) {
            a = red[tid * 3 + 0];
            c = red[tid * 3 + 1];
            d = red[tid * 3 + 2];
        }
#pragma unroll
        for (int off = NWAVES / 2; off > 0; off >>= 1) {
            a += __shfl_xor(a, off, 32);
            c += __shfl_xor(c, off, 32);
            d += __shfl_xor(d, off, 32);
        }
        if (tid == 0) {
            float* o = ws + ((size_t)b * SEGS + s) * 3;
            o[0] = a;   // pt
            o[1] = c;   // tt
            o[2] = d;   // nn
        }
    }
}

// ---------------------------------------------------------------------------
// Kernel 2: one wave32. Each lane finishes 2 rows (rows lane and lane+32):
// sums the 16 segment partials, applies scale/snr/log10. The resulting 64
// per-row values live in a 2-VGPR pair across 32 lanes == exactly the
// A-matrix of V_WMMA_F32_16X16X4_F32 (16x4 fp32). With B = all-ones,
// D[m,n] = sum_k A[m,k], so summing the 8 D VGPRs in-lane gives the sum of
// 8 A-rows (rows 0-7 in lanes 0-15, rows 8-15 in lanes 16-31); one
// shfl_xor(16) add completes the 64-value sum. EXEC is all ones (32 threads,
// uniform control flow) as WMMA requires.
// ---------------------------------------------------------------------------
__global__ __launch_bounds__(32)
void snr_final_kernel(const float* __restrict__ ws, float* __restrict__ out)
{
    const int lane = threadIdx.x;

    float res2[2];
#pragma unroll
    for (int j = 0; j < 2; ++j) {
        const int r = lane + j * 32;
        float pt = 0.0f, tt = 0.0f, nn = 0.0f;
#pragma unroll
        for (int s = 0; s < SEGS; ++s) {
            const float* q = ws + ((size_t)r * SEGS + s) * 3;
            pt += q[0];
            tt += q[1];
            nn += q[2];
        }
        const float scale = pt / (tt + EPS_F);
        const float st_st = scale * scale * tt;
        const float snr   = nn / (st_st + EPS_F);
        res2[j] = 10.0f * log10f(snr);
    }

    v2f a;    a.x = res2[0]; a.y = res2[1];
    v2f ones; ones.x = 1.0f; ones.y = 1.0f;
    v8f cacc = {};
    // 8 args: (neg_a, A, neg_b, B, c_mod, C, reuse_a, reuse_b)
    v8f d = __builtin_amdgcn_wmma_f32_16x16x4_f32(
        false, a, false, ones, (short)0, cacc, false, false);

    float tot = d[0] + d[1] + d[2] + d[3] + d[4] + d[5] + d[6] + d[7];
    tot += __shfl_xor(tot, 16, 32);   // combine rows 0-7 with rows 8-15

    if (lane == 0) {
        out[0] = tot * (1.0f / (float)B_ROWS);
    }
}

// ---------------------------------------------------------------------------
extern "C" void kernel_launch(void* const* d_in, const int* in_sizes, int n_in,
                              void* d_out, int out_size, void* d_ws, size_t ws_size,
                              hipStream_t stream)
{
    const float* target     = (const float*)d_in[0];
    const float* prediction = (const float*)d_in[1];
    const float* noise      = (const float*)d_in[2];
    float*       out        = (float*)d_out;
    float*       ws         = (float*)d_ws;   // needs 64*16*3*4 = 12 KB

    dim3 grid(SEGS, B_ROWS);                  // 1024 blocks x 8 waves
    snr_partial_kernel<<<grid, TPB, 0, stream>>>(target, prediction, noise, ws);
    snr_final_kernel<<<1, 32, 0, stream>>>(ws, out);
}